// SpatialTemporal2d_58368605552853
// MI455X (gfx1250) — compile-verified
//
#include <hip/hip_runtime.h>
#include <hip/hip_bf16.h>
#include <math.h>

typedef float v2f __attribute__((ext_vector_type(2)));
typedef float v8f __attribute__((ext_vector_type(8)));

#define BIG (1 << 30)
#define PI2 6.283185307179586f

// ---------------- workspace layout (float offsets) ----------------
#define OFF_BXF   0         // [64][32]  forward x/y DFT (16 modes, re/-sin)
#define OFF_B2Z   2048      // [80][32]  forward z (complex-in) 16 modes {0..7,13..20}
#define OFF_B2Y   4608      // [128][64] forward y (xy branch)  32 modes {0..15,17..32}
#define OFF_B4Z   12800     // [32][80]  hermitian inverse z
#define OFF_B4Y   15360     // [64][128] hermitian inverse y
#define OFF_B5    23552     // [32][64]  complex inverse (len 64) with 1/64
#define OFF_WC    25600     // [32][32]  conv weight (i,o)
#define OFF_BF0   26624     // [32][64]  FF w0 (o,hid)
#define OFF_BF1   28672     // [64][32]  FF w1 (hid,o)
#define OFF_B3    30720     // up to [512][64][64] mixing blocks
#define OFF_G1    2127872   // 10485760: forward-x (or -y) result
#define OFF_F     12613632  // 5242880 : after 2nd forward transform
#define OFF_O     17856512  // 5242880 : after channel mix
#define OFF_Q     23099392  // 10485760: after first inverse transform
// total 33585152 floats = 134.3 MB

struct GemmDesc {
  int M, N, K, tilesM;
  int gA, gB, gC;                  // per-blockIdx.y base strides
  int Dm0, Dm1, Dm2;               // row-index decomposition divisors
  int sAm0, sAm1, sAm2, sAm3;      // A row strides per level
  int sCm0, sCm1, sCm2, sCm3;      // C row strides per level
  int kd;                          // 1 if k = (k1, ri) with Dk0==2
  int sAk0, sAk1;                  // A k strides (ri, major)
  int Dn0, sCn0, sCn1;             // C col decomposition
  int flags;                       // 1=accumulate, 2=bias, 4=relu
};

__device__ __forceinline__ int mdec(int m, int D0, int D1, int D2,
                                    int s0, int s1, int s2, int s3) {
  int m0 = m % D0; m /= D0;
  int m1 = m % D1; m /= D1;
  int m2 = m % D2; m /= D2;
  return m0 * s0 + m1 * s1 + m2 * s2 + m * s3;
}

// Generic fp32 WMMA GEMM: one 16x16 C tile per wave32, K-loop of 16x16x4 steps.
__global__ __launch_bounds__(32) void gemm_wmma(const float* __restrict__ A,
                                                const float* __restrict__ B,
                                                float* __restrict__ C,
                                                const float* __restrict__ bias,
                                                GemmDesc d) {
  int lane = threadIdx.x;
  int l = lane & 15, half = lane >> 4;
  int tm = blockIdx.x % d.tilesM;
  int tn = blockIdx.x / d.tilesM;
  int g = blockIdx.y;
  A += (size_t)g * d.gA;
  B += (size_t)g * d.gB;
  C += (size_t)g * d.gC;

  int ma = tm * 16 + l;
  int aOff = mdec(ma, d.Dm0, d.Dm1, d.Dm2, d.sAm0, d.sAm1, d.sAm2, d.sAm3);
  int nb = tn * 16 + l;
  int inc = d.kd ? d.sAk0 : d.sAk1;

  v8f acc = {};
  for (int ks = 0; ks < d.K; ks += 4) {
    int kk = ks + half * 2;                 // even -> ri part is 0
    int base = aOff + (kk >> d.kd) * d.sAk1;
    v2f a; a.x = A[base]; a.y = A[base + inc];
    v2f b; b.x = B[kk * d.N + nb]; b.y = B[(kk + 1) * d.N + nb];
    acc = __builtin_amdgcn_wmma_f32_16x16x4_f32(false, a, false, b,
                                                (short)0, acc, false, false);
  }
#pragma unroll
  for (int r = 0; r < 8; ++r) {
    int m = tm * 16 + half * 8 + r;
    int n = tn * 16 + l;
    int cOff = mdec(m, d.Dm0, d.Dm1, d.Dm2, d.sCm0, d.sCm1, d.sCm2, d.sCm3)
             + (n % d.Dn0) * d.sCn0 + (n / d.Dn0) * d.sCn1;
    float v = acc[r];
    if (d.flags & 2) v += bias[n];
    if (d.flags & 4) v = fmaxf(v, 0.0f);
    if (d.flags & 1) v += C[cOff];
    C[cOff] = v;
  }
}

// Build all small transform matrices + repacked dense weights.
__global__ void pack_small(float* __restrict__ W, const float* __restrict__ wW,
                           const float* __restrict__ w0, const float* __restrict__ w1) {
  int t = blockIdx.x * blockDim.x + threadIdx.x;
  if (t >= 30720) return;
  int u = t;
  if (u < 2048) {                    // BXF [64][32]: e^{-i 2pi k x/64}
    int x = u >> 5, n = u & 31, k = n >> 1, r = n & 1;
    float th = PI2 * (float)((k * x) & 63) / 64.0f;
    W[OFF_BXF + u] = r ? -sinf(th) : cosf(th);
    return;
  } u -= 2048;
  if (u < 2560) {                    // B2Z [80][32] rows (z,ri), cols (j,ro)
    int row = u / 32, col = u % 32;
    int z = row >> 1, ri = row & 1, j = col >> 1, ro = col & 1;
    int kz = (j < 8) ? j : (j + 5);
    float th = PI2 * (float)((kz * z) % 40) / 40.0f;
    float c = cosf(th), s = sinf(th);
    W[OFF_B2Z + u] = (ri == ro) ? c : (ri ? s : -s);
    return;
  } u -= 2560;
  if (u < 8192) {                    // B2Y [128][64]
    int row = u / 64, col = u % 64;
    int y = row >> 1, ri = row & 1, j = col >> 1, ro = col & 1;
    int ky = (j < 16) ? j : (j + 1);
    float th = PI2 * (float)((ky * y) & 63) / 64.0f;
    float c = cosf(th), s = sinf(th);
    W[OFF_B2Y + u] = (ri == ro) ? c : (ri ? s : -s);
    return;
  } u -= 8192;
  if (u < 2560) {                    // B4Z [32][80] rows (j,ri), cols (z,qr)
    int row = u / 80, col = u % 80;
    int j = row >> 1, ri = row & 1, z = col >> 1, qr = col & 1;
    int kz = (j < 8) ? j : (j + 5);
    float w = (kz == 0 || kz == 20) ? 1.0f : 2.0f;
    float th = PI2 * (float)((kz * z) % 40) / 40.0f;
    float a = w * cosf(th) / 40.0f, bb = -w * sinf(th) / 40.0f;
    W[OFF_B4Z + u] = (ri == qr) ? a : (ri ? bb : -bb);
    return;
  } u -= 2560;
  if (u < 8192) {                    // B4Y [64][128]
    int row = u / 128, col = u % 128;
    int j = row >> 1, ri = row & 1, y = col >> 1, qr = col & 1;
    int ky = (j < 16) ? j : (j + 1);
    float w = (ky == 0 || ky == 32) ? 1.0f : 2.0f;
    float th = PI2 * (float)((ky * y) & 63) / 64.0f;
    float a = w * cosf(th) / 64.0f, bb = -w * sinf(th) / 64.0f;
    W[OFF_B4Y + u] = (ri == qr) ? a : (ri ? bb : -bb);
    return;
  } u -= 8192;
  if (u < 2048) {                    // B5 [32][64]: (1/64) e^{+i 2pi k x/64}
    int row = u / 64, x = u % 64;
    int kx = row >> 1, ri = row & 1;
    float th = PI2 * (float)((kx * x) & 63) / 64.0f;
    W[OFF_B5 + u] = ri ? (-sinf(th) / 64.0f) : (cosf(th) / 64.0f);
    return;
  } u -= 2048;
  if (u < 1024) {                    // WC [i][o] = w_weight[o][i]
    int i = u / 32, o = u % 32;
    W[OFF_WC + u] = wW[o * 32 + i];
    return;
  } u -= 1024;
  if (u < 2048) {                    // BF0 [o][hid] = ff_w0[hid][o]
    int o = u / 64, h = u % 64;
    W[OFF_BF0 + u] = w0[h * 32 + o];
    return;
  } u -= 2048;
  { // BF1 [hid][o] = ff_w1[o][hid]
    int h = u / 32, o = u % 32;
    W[OFF_BF1 + u] = w1[o * 64 + h];
  }
}

// Complex channel-mix weights -> per-(mode,mode) 64x64 real blocks.
__global__ void pack_b3(float* __restrict__ dst, const float* __restrict__ fw,
                        const float* __restrict__ fw2, int nG, int J, int MB) {
  int t = blockIdx.x * blockDim.x + threadIdx.x;
  if (t >= nG * 4096) return;
  int g = t >> 12, rem = t & 4095;
  int ka = g / J, j = g % J;
  int row = rem >> 6, col = rem & 63;
  int i = row >> 1, ri = row & 1, o = col >> 1, ro = col & 1;
  const float* w; int jl;
  if (j < MB) { w = fw; jl = j; } else { w = fw2; jl = j - MB; }
  int idx = ((((i * 32 + o) * 16 + ka) * MB) + jl) * 2;
  float wre = w[idx], wim = w[idx + 1];
  dst[t] = (ri == ro) ? wre : (ri ? -wim : wim);
}

// Fused FFN: h -> relu(h@W0^T+b0) -> @W1^T+b1, hidden held in LDS. In place on out.
__global__ __launch_bounds__(32) void ff_fused(const float* __restrict__ h,
                                               const float* __restrict__ B0,
                                               const float* __restrict__ b0,
                                               const float* __restrict__ B1,
                                               const float* __restrict__ b1,
                                               float* __restrict__ out) {
  __shared__ float lds[16 * 64];
  int lane = threadIdx.x, l = lane & 15, half = lane >> 4;
  int row0 = blockIdx.x * 16;
  const float* Ar = h + (size_t)(row0 + l) * 32;
#pragma unroll
  for (int nt = 0; nt < 4; ++nt) {
    v8f acc = {};
#pragma unroll
    for (int ks = 0; ks < 32; ks += 4) {
      int kk = ks + half * 2;
      v2f a; a.x = Ar[kk]; a.y = Ar[kk + 1];
      v2f b; b.x = B0[kk * 64 + nt * 16 + l]; b.y = B0[(kk + 1) * 64 + nt * 16 + l];
      acc = __builtin_amdgcn_wmma_f32_16x16x4_f32(false, a, false, b,
                                                  (short)0, acc, false, false);
    }
#pragma unroll
    for (int r = 0; r < 8; ++r)
      lds[(r + half * 8) * 64 + nt * 16 + l] = fmaxf(acc[r] + b0[nt * 16 + l], 0.0f);
  }
  __syncthreads();
#pragma unroll
  for (int nt = 0; nt < 2; ++nt) {
    v8f acc = {};
#pragma unroll
    for (int ks = 0; ks < 64; ks += 4) {
      int kk = ks + half * 2;
      v2f a; a.x = lds[l * 64 + kk]; a.y = lds[l * 64 + kk + 1];
      v2f b; b.x = B1[kk * 32 + nt * 16 + l]; b.y = B1[(kk + 1) * 32 + nt * 16 + l];
      acc = __builtin_amdgcn_wmma_f32_16x16x4_f32(false, a, false, b,
                                                  (short)0, acc, false, false);
    }
#pragma unroll
    for (int r = 0; r < 8; ++r)
      out[(size_t)(row0 + r + half * 8) * 32 + nt * 16 + l] = acc[r] + b1[nt * 16 + l];
  }
}

static GemmDesc base_desc(int M, int N, int K) {
  GemmDesc d{};
  d.M = M; d.N = N; d.K = K; d.tilesM = M / 16;
  d.Dm0 = BIG; d.Dm1 = 1; d.Dm2 = 1;
  d.Dn0 = BIG;
  return d;
}

extern "C" void kernel_launch(void* const* d_in, const int* in_sizes, int n_in,
                              void* d_out, int out_size, void* d_ws, size_t ws_size,
                              hipStream_t stream) {
  (void)in_sizes; (void)n_in; (void)out_size; (void)ws_size;
  const float* x      = (const float*)d_in[0];
  const float* wW     = (const float*)d_in[1];
  const float* wB     = (const float*)d_in[2];
  const float* fw_xy  = (const float*)d_in[3];
  const float* fw_yz  = (const float*)d_in[4];
  const float* fw_xz  = (const float*)d_in[5];
  const float* fw2_xy = (const float*)d_in[6];
  const float* fw2_yz = (const float*)d_in[7];
  const float* fw2_xz = (const float*)d_in[8];
  const float* ffw0   = (const float*)d_in[9];
  const float* ffb0   = (const float*)d_in[10];
  const float* ffw1   = (const float*)d_in[11];
  const float* ffb1   = (const float*)d_in[12];
  float* out = (float*)d_out;
  float* W   = (float*)d_ws;

  auto G = [&](const float* A, const float* B, float* C, const float* bias,
               const GemmDesc& d, int gy) {
    dim3 grid((unsigned)(d.tilesM * (d.N / 16)), (unsigned)gy);
    gemm_wmma<<<grid, dim3(32), 0, stream>>>(A, B, C, bias, d);
  };

  pack_small<<<120, 256, 0, stream>>>(W, wW, ffw0, ffw1);

  // ---- 1x1 conv branch: out = x @ Wc + bias (beta=0 init of accumulator) ----
  {
    GemmDesc d = base_desc(655360, 32, 32);
    d.sAm0 = 32; d.sAk1 = 1; d.sCm0 = 32; d.sCn0 = 1; d.flags = 2;
    G(x, W + OFF_WC, out, wB, d, 1);
  }

  // ---- shared forward-x DFT: G1[b][y][z][i][kx][ri] ----
  {
    GemmDesc d = base_desc(327680, 32, 64);
    d.Dm0 = 81920; d.Dm1 = BIG;
    d.sAm0 = 1; d.sAm1 = 5242880;
    d.sCm0 = 32; d.sCm1 = 2621440;
    d.sAk1 = 81920; d.sCn0 = 1;
    G(x, W + OFF_BXF, W + OFF_G1, nullptr, d, 1);
  }

  // ================= xz branch =================
  pack_b3<<<(256 * 4096) / 256, 256, 0, stream>>>(W + OFF_B3, fw_xz, fw2_xz, 256, 16, 8);
  GemmDesc s2 = base_desc(512, 32, 80);        // forward z per (b,y)
  s2.gA = 40960; s2.gC = 64;
  s2.Dm0 = 16; s2.Dm1 = BIG;
  s2.sAm0 = 2; s2.sAm1 = 32;
  s2.sCm0 = 262144; s2.sCm1 = 2;
  s2.kd = 1; s2.sAk0 = 1; s2.sAk1 = 1024;
  s2.Dn0 = 2; s2.sCn0 = 1; s2.sCn1 = 16384;
  G(W + OFF_G1, W + OFF_B2Z, W + OFF_F, nullptr, s2, 256);

  GemmDesc s3 = base_desc(256, 64, 64);        // channel mix per (kx,jz)
  s3.gA = 16384; s3.gB = 4096; s3.gC = 2;
  s3.sAm0 = 64; s3.sAk1 = 1; s3.sCm0 = 16384;
  s3.Dn0 = 2; s3.sCn0 = 1; s3.sCn1 = 512;
  G(W + OFF_F, W + OFF_B3, W + OFF_O, nullptr, s3, 256);

  GemmDesc s4 = base_desc(131072, 80, 32);     // hermitian inverse z
  s4.Dm0 = 16; s4.Dm1 = BIG;
  s4.sAm0 = 32; s4.sAm1 = 512;
  s4.sCm0 = 2; s4.sCm1 = 1280;
  s4.sAk1 = 1; s4.Dn0 = 2; s4.sCn0 = 1; s4.sCn1 = 32;
  G(W + OFF_O, W + OFF_B4Z, W + OFF_Q, nullptr, s4, 1);

  {                                            // complex inverse x, accumulate
    GemmDesc d = base_desc(327680, 64, 32);
    d.Dm0 = 40; d.Dm1 = 32; d.Dm2 = 64;
    d.sAm0 = 32; d.sAm1 = 1280; d.sAm2 = 40960; d.sAm3 = 2621440;
    d.sCm0 = 32; d.sCm1 = 1; d.sCm2 = 1280; d.sCm3 = 5242880;
    d.sAk1 = 1; d.sCn0 = 81920; d.flags = 1;
    G(W + OFF_Q, W + OFF_B5, out, nullptr, d, 1);
  }

  // ================= xy branch (reuses G1) =================
  pack_b3<<<(512 * 4096) / 256, 256, 0, stream>>>(W + OFF_B3, fw_xy, fw2_xy, 512, 32, 16);
  {                                            // forward y per b
    GemmDesc d = base_desc(20480, 64, 128);
    d.gA = 2621440; d.gC = 2560;
    d.Dm0 = 16; d.Dm1 = 32; d.Dm2 = BIG;
    d.sAm0 = 2; d.sAm1 = 32; d.sAm2 = 1024;
    d.sCm0 = 327680; d.sCm1 = 2; d.sCm2 = 64;
    d.kd = 1; d.sAk0 = 1; d.sAk1 = 40960;
    d.Dn0 = 2; d.sCn0 = 1; d.sCn1 = 10240;
    G(W + OFF_G1, W + OFF_B2Y, W + OFF_F, nullptr, d, 4);
  }
  {                                            // channel mix per (kx,jy)
    GemmDesc d = base_desc(160, 64, 64);
    d.gA = 10240; d.gB = 4096; d.gC = 2;
    d.sAm0 = 64; d.sAk1 = 1; d.sCm0 = 32768;
    d.Dn0 = 2; d.sCn0 = 1; d.sCn1 = 1024;
    G(W + OFF_F, W + OFF_B3, W + OFF_O, nullptr, d, 512);
  }
  {                                            // hermitian inverse y
    GemmDesc d = base_desc(81920, 128, 64);
    d.Dm0 = 16; d.Dm1 = 32; d.Dm2 = BIG;
    d.sAm0 = 64; d.sAm1 = 1024; d.sAm2 = 32768;
    d.sCm0 = 2; d.sCm1 = 32; d.sCm2 = 65536;
    d.sAk1 = 1; d.Dn0 = 2; d.sCn0 = 1; d.sCn1 = 1024;
    G(W + OFF_O, W + OFF_B4Y, W + OFF_Q, nullptr, d, 1);
  }
  {                                            // complex inverse x, accumulate
    GemmDesc d = base_desc(327680, 64, 32);
    d.Dm0 = 32; d.Dm1 = 64; d.Dm2 = 40;
    d.sAm0 = 32; d.sAm1 = 1024; d.sAm2 = 65536; d.sAm3 = 2621440;
    d.sCm0 = 1; d.sCm1 = 1280; d.sCm2 = 32; d.sCm3 = 5242880;
    d.sAk1 = 1; d.sCn0 = 81920; d.flags = 1;
    G(W + OFF_Q, W + OFF_B5, out, nullptr, d, 1);
  }

  // ================= yz branch =================
  pack_b3<<<(256 * 4096) / 256, 256, 0, stream>>>(W + OFF_B3, fw_yz, fw2_yz, 256, 16, 8);
  {                                            // forward y DFT: G1[b][x][z][i][ky][ri]
    GemmDesc d = base_desc(327680, 32, 64);
    d.Dm0 = 1280; d.Dm1 = 64; d.Dm2 = BIG;
    d.sAm0 = 1; d.sAm1 = 81920; d.sAm2 = 5242880;
    d.sCm0 = 32; d.sCm1 = 40960; d.sCm2 = 2621440;
    d.sAk1 = 1280; d.sCn0 = 1;
    G(x, W + OFF_BXF, W + OFF_G1, nullptr, d, 1);
  }
  G(W + OFF_G1, W + OFF_B2Z, W + OFF_F, nullptr, s2, 256);  // forward z (same desc)
  G(W + OFF_F, W + OFF_B3, W + OFF_O, nullptr, s3, 256);    // channel mix
  G(W + OFF_O, W + OFF_B4Z, W + OFF_Q, nullptr, s4, 1);     // hermitian inverse z
  {                                            // complex inverse y, accumulate
    GemmDesc d = base_desc(327680, 64, 32);
    d.Dm0 = 40; d.Dm1 = 32; d.Dm2 = 64;
    d.sAm0 = 32; d.sAm1 = 1280; d.sAm2 = 40960; d.sAm3 = 2621440;
    d.sCm0 = 32; d.sCm1 = 1; d.sCm2 = 81920; d.sCm3 = 5242880;
    d.sAk1 = 1; d.sCn0 = 1280; d.flags = 1;
    G(W + OFF_Q, W + OFF_B5, out, nullptr, d, 1);
  }

  // ---- fused FFN, in place on out ----
  ff_fused<<<40960, 32, 0, stream>>>(out, W + OFF_BF0, ffb0, W + OFF_BF1, ffb1, out);
}